// Gat_64433099375269
// MI455X (gfx1250) — compile-verified
//
#include <hip/hip_runtime.h>
#include <math.h>

#define NN      100000
#define NE      3200000
#define ETOT    (NE + NN)        // edges + self loops
#define NFEAT   512
#define HDIM    64
#define NCLASS  40
#define NCPAD   48               // NCLASS padded to 3 col tiles of 16
#define NLAYERS 4
#define NEG_SLOPE 0.2f

typedef __attribute__((ext_vector_type(2))) float v2f;
typedef __attribute__((ext_vector_type(8))) float v8f;

// ---------------------------------------------------------------------------
// float atomic max via sign-split (exact for mixed signs; m initialized -inf)
// ---------------------------------------------------------------------------
__device__ __forceinline__ void atomic_max_f32(float* addr, float val) {
    if (val >= 0.0f) atomicMax((int*)addr, __float_as_int(val));
    else             atomicMin((unsigned int*)addr, __float_as_uint(val));
}

__device__ __forceinline__ float lrelu(float v) {
    return v > 0.0f ? v : NEG_SLOPE * v;
}

// ---------------------------------------------------------------------------
// WMMA f32 GEMM: C[M x HDIM] = A[M x K] * B[K x HDIM] (+bias), row-major.
// One wave computes one 16x16 output tile via V_WMMA_F32_16X16X4_F32.
// Block = 128 threads = 4 waves = one 16-row stripe covering all 64 columns.
// Grid = M/16 (M = 100000 = 6250*16, exact).
//
// ISA VGPR layouts (CDNA5 ISA §7.12.2):
//   A 16x4 f32 : lane m=lane&15, kpair=(lane>>4)*2 -> {A[m][k+kp], A[m][k+kp+1]}
//   B  4x16 f32: lane n=lane&15, kpair=(lane>>4)*2 -> {B[k+kp][n], B[k+kp+1][n]}
//   C 16x16 f32: vgpr i, lane l -> C[i + (l>=16?8:0)][l&15]
// ---------------------------------------------------------------------------
template<int K, int LDA>
__global__ void k_gemm_tile(const float* __restrict__ A,
                            const float* __restrict__ B,
                            const float* __restrict__ bias,
                            float* __restrict__ C) {
    const int tileM   = blockIdx.x;
    const int wave    = threadIdx.x >> 5;
    const int lane    = threadIdx.x & 31;
    const int l15     = lane & 15;
    const int kpb     = (lane >> 4) << 1;      // 0 or 2
    const int colbase = wave * 16;

    const float* ap = A + (size_t)(tileM * 16 + l15) * LDA + kpb;
    const float* bp = B + (size_t)kpb * HDIM + colbase + l15;

    v8f c = {};
#pragma unroll 4
    for (int k = 0; k < K; k += 4) {
        v2f a, b;
        a.x = ap[0];
        a.y = ap[1];
        b.x = bp[0];
        b.y = bp[HDIM];
        ap += 4;
        bp += 4 * HDIM;
        c = __builtin_amdgcn_wmma_f32_16x16x4_f32(
                false, a, false, b, (short)0, c, false, false);
    }

    const float bv   = bias ? bias[colbase + l15] : 0.0f;
    const int mbase  = tileM * 16 + ((lane >> 4) << 3);
    float* cp = C + (size_t)mbase * HDIM + colbase + l15;
#pragma unroll
    for (int i = 0; i < 8; ++i)
        cp[(size_t)i * HDIM] = c[i] + bv;
}

// ---------------------------------------------------------------------------
// Output GEMM: out[N x 40] = h[N x 64] * W_out[64 x 40] + b_out.
// W_out staged once into zero-padded 64x48 LDS -> branch-free K loop
// (avoids per-load exec-mask save/restore seen in round-1 asm).
// Block = 96 threads = 3 waves (3 column tiles).
// ---------------------------------------------------------------------------
__global__ void k_gemm_out(const float* __restrict__ A,
                           const float* __restrict__ B,
                           const float* __restrict__ bias,
                           float* __restrict__ C) {
    __shared__ float Bs[HDIM * NCPAD];          // 12 KB of the 320 KB/WGP LDS

    for (int i = threadIdx.x; i < HDIM * NCPAD; i += 96) {
        const int kk = i / NCPAD, cc = i - kk * NCPAD;
        Bs[i] = (cc < NCLASS) ? B[(size_t)kk * NCLASS + cc] : 0.0f;
    }
    __syncthreads();

    const int tileM  = blockIdx.x;
    const int wave   = threadIdx.x >> 5;
    const int lane   = threadIdx.x & 31;
    const int l15    = lane & 15;
    const int kpb    = (lane >> 4) << 1;
    const int col    = wave * 16 + l15;         // 0..47

    const float* ap = A + (size_t)(tileM * 16 + l15) * HDIM + kpb;
    const float* bp = Bs + kpb * NCPAD + col;

    v8f c = {};
#pragma unroll
    for (int k = 0; k < HDIM; k += 4) {
        v2f a, b;
        a.x = ap[0];
        a.y = ap[1];
        b.x = bp[0];
        b.y = bp[NCPAD];
        ap += 4;
        bp += 4 * NCPAD;
        c = __builtin_amdgcn_wmma_f32_16x16x4_f32(
                false, a, false, b, (short)0, c, false, false);
    }

    if (col < NCLASS) {
        const float bv  = bias[col];
        const int mbase = tileM * 16 + ((lane >> 4) << 3);
#pragma unroll
        for (int i = 0; i < 8; ++i)
            C[(size_t)(mbase + i) * NCLASS + col] = c[i] + bv;
    }
}

// ---------------------------------------------------------------------------
// One-time: materialize int32 src/dst (with self loops appended).
// ---------------------------------------------------------------------------
__global__ void k_edge_prep(const long long* __restrict__ ei,
                            int* __restrict__ s32,
                            int* __restrict__ d32) {
    const int e = blockIdx.x * blockDim.x + threadIdx.x;
    if (e >= ETOT) return;
    if (e < NE) { s32[e] = (int)ei[e]; d32[e] = (int)ei[e + NE]; }
    else        { s32[e] = d32[e] = e - NE; }
}

// ---------------------------------------------------------------------------
// Per-node prep: attention logit dots + init segment buffers + zero acc.
// ---------------------------------------------------------------------------
__global__ void k_node_prep(const float* __restrict__ hw,
                            const float* __restrict__ att_s,
                            const float* __restrict__ att_d,
                            float* __restrict__ a_src,
                            float* __restrict__ a_dst,
                            float* __restrict__ m,
                            float* __restrict__ denom,
                            float* __restrict__ acc) {
    const int n = blockIdx.x * blockDim.x + threadIdx.x;
    if (n >= NN) return;
    const float4* hv = (const float4*)(hw + (size_t)n * HDIM);
    const float4* ws = (const float4*)att_s;
    const float4* wd = (const float4*)att_d;
    float ss = 0.0f, sd = 0.0f;
    float4* av = (float4*)(acc + (size_t)n * HDIM);
    const float4 z = make_float4(0.0f, 0.0f, 0.0f, 0.0f);
#pragma unroll
    for (int i = 0; i < HDIM / 4; ++i) {
        const float4 h4 = hv[i];
        const float4 s4 = ws[i];
        const float4 d4 = wd[i];
        ss += h4.x * s4.x + h4.y * s4.y + h4.z * s4.z + h4.w * s4.w;
        sd += h4.x * d4.x + h4.y * d4.y + h4.z * d4.z + h4.w * d4.w;
        av[i] = z;
    }
    a_src[n] = ss;
    a_dst[n] = sd;
    m[n]     = -INFINITY;
    denom[n] = 0.0f;
}

// Pass A: ev[e] = leaky_relu(logit); segment max into m[dst].
__global__ void k_edge_logits(const int* __restrict__ s32,
                              const int* __restrict__ d32,
                              const float* __restrict__ a_src,
                              const float* __restrict__ a_dst,
                              float* __restrict__ ev,
                              float* __restrict__ m) {
    const int e = blockIdx.x * blockDim.x + threadIdx.x;
    if (e >= ETOT) return;
    const int d = d32[e];
    const float v = lrelu(a_src[s32[e]] + a_dst[d]);
    ev[e] = v;
    atomic_max_f32(&m[d], v);
}

// Pass B: ev[e] <- exp(ev[e]-m[d]); segment sum into denom[dst].
__global__ void k_edge_expsum(const int* __restrict__ d32,
                              float* __restrict__ ev,
                              const float* __restrict__ m,
                              float* __restrict__ denom) {
    const int e = blockIdx.x * blockDim.x + threadIdx.x;
    if (e >= ETOT) return;
    const int d = d32[e];
    const float ex = __expf(ev[e] - m[d]);
    ev[e] = ex;                      // in place; per-edge slot, no conflict
    atomicAdd(&denom[d], ex);
}

// Pass C (heavy): 16 lanes per edge; float4 gather of hW[src], 4 atomics
// into the L2-resident accumulator. alpha from linear ev[e] + denom[d]
// (16 consecutive lanes hit the same cachelines).
__global__ void k_edge_scatter(const int* __restrict__ s32,
                               const int* __restrict__ d32,
                               const float* __restrict__ ev,
                               const float* __restrict__ denom,
                               const float* __restrict__ hw,
                               float* __restrict__ acc) {
    const long long t = (long long)blockIdx.x * blockDim.x + threadIdx.x;
    const int e = (int)(t >> 4);
    if (e >= ETOT) return;
    const int q = (int)(t & 15);
    const int s = s32[e];
    const int d = d32[e];
    const float alpha = ev[e] / denom[d];
    const float4 h4   = ((const float4*)(hw + (size_t)s * HDIM))[q];
    float* op = acc + (size_t)d * HDIM + q * 4;
    atomicAdd(op + 0, alpha * h4.x);
    atomicAdd(op + 1, alpha * h4.y);
    atomicAdd(op + 2, alpha * h4.z);
    atomicAdd(op + 3, alpha * h4.w);
}

// h = h + elu(acc + b_conv[l])   (residual, in place)
__global__ void k_finalize(float* __restrict__ h,
                           const float* __restrict__ acc,
                           const float* __restrict__ bias) {
    const long long t = (long long)blockIdx.x * blockDim.x + threadIdx.x;
    const int n = (int)(t >> 4);
    if (n >= NN) return;
    const int q = (int)(t & 15);
    const float4 a4 = ((const float4*)(acc + (size_t)n * HDIM))[q];
    const float4 b4 = ((const float4*)bias)[q];
    float4* hp = (float4*)(h + (size_t)n * HDIM) + q;
    float4 h4 = *hp;
    float z;
    z = a4.x + b4.x; h4.x += (z > 0.0f ? z : __expf(z) - 1.0f);
    z = a4.y + b4.y; h4.y += (z > 0.0f ? z : __expf(z) - 1.0f);
    z = a4.z + b4.z; h4.z += (z > 0.0f ? z : __expf(z) - 1.0f);
    z = a4.w + b4.w; h4.w += (z > 0.0f ? z : __expf(z) - 1.0f);
    *hp = h4;
}

// ---------------------------------------------------------------------------
extern "C" void kernel_launch(void* const* d_in, const int* in_sizes, int n_in,
                              void* d_out, int out_size, void* d_ws, size_t ws_size,
                              hipStream_t stream) {
    (void)in_sizes; (void)n_in; (void)out_size; (void)ws_size;

    const float*     x       = (const float*)d_in[0];
    const long long* ei      = (const long long*)d_in[1];   // int64 [2, NE]
    const float*     W_in    = (const float*)d_in[2];
    const float*     b_in    = (const float*)d_in[3];
    const float*     W_conv  = (const float*)d_in[4];       // [L,64,64]
    const float*     att_src = (const float*)d_in[5];       // [L,64]
    const float*     att_dst = (const float*)d_in[6];       // [L,64]
    const float*     b_conv  = (const float*)d_in[7];       // [L,64]
    const float*     W_out   = (const float*)d_in[8];
    const float*     b_out   = (const float*)d_in[9];
    float*           out     = (float*)d_out;

    // workspace (all 4-byte elems):
    // h | hw | acc | a_src | a_dst | m | denom | s32 | d32 | ev   (~118 MB)
    float* h    = (float*)d_ws;
    float* hw   = h    + (size_t)NN * HDIM;
    float* acc  = hw   + (size_t)NN * HDIM;
    float* asrc = acc  + (size_t)NN * HDIM;
    float* adst = asrc + NN;
    float* mbuf = adst + NN;
    float* dbuf = mbuf + NN;
    int*   s32  = (int*)(dbuf + NN);
    int*   d32  = s32  + (size_t)ETOT;
    float* ev   = (float*)(d32 + (size_t)ETOT);

    const int ROW_TILES = NN / 16;                 // 6250 exact
    const int NODE_BLK  = (NN + 255) / 256;
    const int EDGE_BLK  = (ETOT + 255) / 256;
    const int SCAT_BLK  = (int)(((long long)ETOT * 16 + 255) / 256);
    const int FIN_BLK   = (int)(((long long)NN * 16 + 255) / 256);

    // one-time int64 -> int32 edge compaction (+self loops)
    k_edge_prep<<<EDGE_BLK, 256, 0, stream>>>(ei, s32, d32);

    // h = x @ W_in + b_in
    k_gemm_tile<NFEAT, NFEAT><<<ROW_TILES, 128, 0, stream>>>(x, W_in, b_in, h);

    for (int l = 0; l < NLAYERS; ++l) {
        const float* Wl = W_conv  + (size_t)l * HDIM * HDIM;
        const float* as = att_src + (size_t)l * HDIM;
        const float* ad = att_dst + (size_t)l * HDIM;
        const float* bc = b_conv  + (size_t)l * HDIM;

        k_gemm_tile<HDIM, HDIM><<<ROW_TILES, 128, 0, stream>>>(h, Wl, nullptr, hw);
        k_node_prep<<<NODE_BLK, 256, 0, stream>>>(hw, as, ad, asrc, adst, mbuf, dbuf, acc);
        k_edge_logits<<<EDGE_BLK, 256, 0, stream>>>(s32, d32, asrc, adst, ev, mbuf);
        k_edge_expsum<<<EDGE_BLK, 256, 0, stream>>>(d32, ev, mbuf, dbuf);
        k_edge_scatter<<<SCAT_BLK, 256, 0, stream>>>(s32, d32, ev, dbuf, hw, acc);
        k_finalize<<<FIN_BLK, 256, 0, stream>>>(h, acc, bc);
    }

    k_gemm_out<<<ROW_TILES, 96, 0, stream>>>(h, W_out, b_out, out);
}